// HebbianLayer_42374147342813
// MI455X (gfx1250) — compile-verified
//
#include <hip/hip_runtime.h>

// ---------------------------------------------------------------------------
// Hebbian/Sanger layer for MI455X (gfx1250, wave32).
//   Phase A: y = W x            via V_WMMA_F32_16X16X4_F32 (f32-exact WMMA)
//   Phase B: y_n = y/||y||      (single-block reduction)
//   Phase C: W_new = W + a*(y x^T - y .* rowcumsum(W .* y))
//            as a 3-pass segmented scan over 16 row blocks so that ~4096
//            waves stream W (L2-resident: 128MB W < 192MB L2) and write the
//            result with non-temporal stores.
// ---------------------------------------------------------------------------

typedef __attribute__((ext_vector_type(2))) float v2f;
typedef __attribute__((ext_vector_type(8))) float v8f;

#define IN_DIM          8192
#define OUT_DIM         4096
#define ROW_BLOCKS      16
#define ROWS_PER_BLOCK  (OUT_DIM / ROW_BLOCKS)   // 256
#define ALPHA_F         0.001f
#define EPS_F           1e-12f

// ---------------------------------------------------------------------------
// Phase A: GEMV via WMMA. One block = 8 waves; block b owns rows [16b,16b+16).
// Each wave covers 1/8 of K with V_WMMA_F32_16X16X4_F32, accumulating a
// 16x16 f32 tile in which every column equals the 16-row y-tile (B operand is
// the x-chunk broadcast to all 16 N columns). Wave partials reduced in LDS.
// ---------------------------------------------------------------------------
__global__ __launch_bounds__(256)
void heb_gemv_wmma(const float* __restrict__ W, const float* __restrict__ x,
                   float* __restrict__ y) {
  __shared__ float part[8][16];
  const int tid  = threadIdx.x;
  const int lane = tid & 31;
  const int wave = tid >> 5;
  const int row0 = blockIdx.x << 4;
  const int m    = lane & 15;
  const int koff = (lane & 16) ? 2 : 0;   // A/B 32-bit layout: hi half-wave = K+2

  const float* __restrict__ wrow = W + (size_t)(row0 + m) * IN_DIM + koff;
  const float* __restrict__ xk   = x + koff;

  v8f c = {};
  const int kBeg = wave * (IN_DIM / 8);
  const int kEnd = kBeg + (IN_DIM / 8);
#pragma unroll 4
  for (int k = kBeg; k < kEnd; k += 4) {
    v2f a, b;
    a.x = wrow[k];     a.y = wrow[k + 1];
    b.x = xk[k];       b.y = xk[k + 1];
    // (neg_a, A, neg_b, B, c_mod, C, reuse_a, reuse_b)
    c = __builtin_amdgcn_wmma_f32_16x16x4_f32(false, a, false, b,
                                              (short)0, c, false, false);
  }

  // Column N=0 of D: lane 0 holds M=0..7 (VGPR 0..7), lane 16 holds M=8..15.
  if (lane == 0) {
#pragma unroll
    for (int i = 0; i < 8; ++i) part[wave][i] = c[i];
  }
  if (lane == 16) {
#pragma unroll
    for (int i = 0; i < 8; ++i) part[wave][8 + i] = c[i];
  }
  __syncthreads();
  if (tid < 16) {
    float s = 0.f;
#pragma unroll
    for (int w = 0; w < 8; ++w) s += part[w][tid];
    y[row0 + tid] = s;
  }
}

// ---------------------------------------------------------------------------
// Phase B: y_n = y / max(||y||, eps). Writes y_n into workspace and into
// d_out[0..4095].
// ---------------------------------------------------------------------------
__global__ __launch_bounds__(256)
void heb_normalize(const float* __restrict__ y, float* __restrict__ yn,
                   float* __restrict__ out_yn) {
  __shared__ float red[256];
  __shared__ float s_inv;
  float s = 0.f;
  for (int i = threadIdx.x; i < OUT_DIM; i += 256) {
    float v = y[i];
    s += v * v;
  }
  red[threadIdx.x] = s;
  __syncthreads();
  for (int off = 128; off > 0; off >>= 1) {
    if (threadIdx.x < off) red[threadIdx.x] += red[threadIdx.x + off];
    __syncthreads();
  }
  if (threadIdx.x == 0) s_inv = 1.0f / fmaxf(sqrtf(red[0]), EPS_F);
  __syncthreads();
  const float inv = s_inv;
  for (int i = threadIdx.x; i < OUT_DIM; i += 256) {
    float v = y[i] * inv;
    yn[i]     = v;
    out_yn[i] = v;
  }
}

// ---------------------------------------------------------------------------
// Phase C1: per row-block column sums  colsum[rb][j] = sum_{r in rb} W[r,j]*yn[r]
// Grid (IN_DIM/256, ROW_BLOCKS); thread = one column; loads coalesced 128B/wave.
// ---------------------------------------------------------------------------
__global__ __launch_bounds__(256)
void heb_colsum(const float* __restrict__ W, const float* __restrict__ yn,
                float* __restrict__ colsum) {
  __shared__ float yl[ROWS_PER_BLOCK];
  const int j  = blockIdx.x * 256 + threadIdx.x;
  const int rb = blockIdx.y;
  const int r0 = rb * ROWS_PER_BLOCK;
  yl[threadIdx.x] = yn[r0 + threadIdx.x];
  __syncthreads();

  const float* __restrict__ wp = W + (size_t)r0 * IN_DIM + j;
  float s = 0.f;
#pragma unroll 8
  for (int r = 0; r < ROWS_PER_BLOCK; ++r) {
    if (r + 64 < ROWS_PER_BLOCK)
      __builtin_prefetch(wp + (size_t)(r + 64) * IN_DIM, 0, 0);
    s += wp[(size_t)r * IN_DIM] * yl[r];
  }
  colsum[(size_t)rb * IN_DIM + j] = s;
}

// ---------------------------------------------------------------------------
// Phase C2: per-column exclusive scan across the 16 row blocks (in place).
// ---------------------------------------------------------------------------
__global__ __launch_bounds__(256)
void heb_scan(float* __restrict__ colsum) {
  const int j = blockIdx.x * 256 + threadIdx.x;
  float s = 0.f;
#pragma unroll
  for (int rb = 0; rb < ROW_BLOCKS; ++rb) {
    const size_t idx = (size_t)rb * IN_DIM + j;
    float v = colsum[idx];
    colsum[idx] = s;      // exclusive prefix for this block
    s += v;
  }
}

// ---------------------------------------------------------------------------
// Phase C3: resume running sum from block offset, emit
//   W_new[r,j] = W[r,j] + a*(yn[r]*x[j] - yn[r]*S[r,j])   (inclusive S)
// W read hits L2 (second pass, W fits in 192MB L2); W_new stored NT so the
// 128MB output stream does not evict W.
// ---------------------------------------------------------------------------
__global__ __launch_bounds__(256)
void heb_update(const float* __restrict__ W, const float* __restrict__ x,
                const float* __restrict__ yn, const float* __restrict__ colpre,
                float* __restrict__ out) {
  __shared__ float yl[ROWS_PER_BLOCK];
  const int j  = blockIdx.x * 256 + threadIdx.x;
  const int rb = blockIdx.y;
  const int r0 = rb * ROWS_PER_BLOCK;
  yl[threadIdx.x] = yn[r0 + threadIdx.x];
  __syncthreads();

  const float xj = x[j];
  float s = colpre[(size_t)rb * IN_DIM + j];
  const float* __restrict__ wp = W + (size_t)r0 * IN_DIM + j;
  float* __restrict__ op = out + (size_t)OUT_DIM + (size_t)r0 * IN_DIM + j;

#pragma unroll 8
  for (int r = 0; r < ROWS_PER_BLOCK; ++r) {
    if (r + 64 < ROWS_PER_BLOCK)
      __builtin_prefetch(wp + (size_t)(r + 64) * IN_DIM, 0, 0);
    const float yr = yl[r];
    const float w  = wp[(size_t)r * IN_DIM];
    s += w * yr;                               // inclusive prefix sum
    const float o = w + ALPHA_F * (yr * xj - yr * s);
    __builtin_nontemporal_store(o, op + (size_t)r * IN_DIM);
  }
}

// ---------------------------------------------------------------------------
extern "C" void kernel_launch(void* const* d_in, const int* in_sizes, int n_in,
                              void* d_out, int out_size, void* d_ws, size_t ws_size,
                              hipStream_t stream) {
  (void)in_sizes; (void)n_in; (void)out_size; (void)ws_size;
  const float* x = (const float*)d_in[0];   // [1, 8192]
  const float* W = (const float*)d_in[1];   // [4096, 8192]
  float* out = (float*)d_out;               // [4096 y_n | 4096*8192 W_new]

  float* y      = (float*)d_ws;             // 4096 floats
  float* yn     = y + OUT_DIM;              // 4096 floats
  float* colsum = yn + OUT_DIM;             // ROW_BLOCKS * IN_DIM floats

  heb_gemv_wmma<<<OUT_DIM / 16, 256, 0, stream>>>(W, x, y);
  heb_normalize<<<1, 256, 0, stream>>>(y, yn, out);

  dim3 gridC(IN_DIM / 256, ROW_BLOCKS);
  heb_colsum<<<gridC, 256, 0, stream>>>(W, yn, colsum);
  heb_scan<<<IN_DIM / 256, 256, 0, stream>>>(colsum);
  heb_update<<<gridC, 256, 0, stream>>>(W, x, yn, colsum, out);
}